// YoloLoss_9534827397913
// MI455X (gfx1250) — compile-verified
//
#include <hip/hip_runtime.h>
#include <hip/hip_bf16.h>

#define NB     16
#define NC     25
#define HWCNT  65536          // 256*256
#define PLANE4 16384          // HWCNT/4
#define EPSF   1e-8f

typedef float v2f __attribute__((ext_vector_type(2)));
typedef float v4f __attribute__((ext_vector_type(4)));
typedef float v8f __attribute__((ext_vector_type(8)));

// Exact fp32 sum of one value per lane across the wave32, using
// V_WMMA_F32_16X16X4_F32 with B = ones:
//   A[m][0] = v[lane m] (lanes 0-15, VGPR0/K=0), A[m][2] = v[lane m+16]
//   (lanes 16-31, VGPR0/K=2), A[m][1] = A[m][3] = 0 (VGPR1).
//   D[m][n] = sum_k A[m][k] = v[m] + v[m+16]  (replicated over n)
// Lane<16 holds D rows 0..7 in d[0..7]; lane>=16 holds rows 8..15.
// So sum(d[0..7]) per half + shfl_xor(16) gives the full 32-lane sum.
__device__ __forceinline__ float wave_sum32(float v) {
    v2f a; a[0] = v;    a[1] = 0.0f;
    v2f b; b[0] = 1.0f; b[1] = 1.0f;
    v8f c = {};
    v8f d = __builtin_amdgcn_wmma_f32_16x16x4_f32(
        /*neg_a=*/false, a, /*neg_b=*/false, b,
        /*c_mod=*/(short)0, c, /*reuse_a=*/false, /*reuse_b=*/false);
    float s = ((d[0] + d[1]) + (d[2] + d[3])) + ((d[4] + d[5]) + (d[6] + d[7]));
    s += __shfl_xor(s, 16, 32);
    return s;
}

__global__ __launch_bounds__(256)
void yolo_main(const float* __restrict__ outputs,
               const float* __restrict__ labels,
               float* __restrict__ ws) {
    const int b    = blockIdx.y;                                  // batch
    const int idx4 = blockIdx.x * blockDim.x + threadIdx.x;       // 0..16383 (float4 granules)
    const size_t base4 = (size_t)b * (NC * PLANE4) + idx4;

    const v4f* o4 = reinterpret_cast<const v4f*>(outputs);
    const v4f* l4 = reinterpret_cast<const v4f*>(labels);

    // ---- channel 0: objectness + tp/fp/fn ----
    v4f x4 = __builtin_nontemporal_load(&o4[base4]);
    v4f y4 = __builtin_nontemporal_load(&l4[base4]);

    float obj = 0.f, tp = 0.f, fp = 0.f, fn = 0.f;
    #pragma unroll
    for (int j = 0; j < 4; ++j) {
        float x = x4[j], y = y4[j];
        float xc = fminf(fmaxf(x,        EPSF), 1.0f - EPSF);
        float x1 = fminf(fmaxf(1.0f - x, EPSF), 1.0f - EPSF);
        obj += -y * (1.0f - xc) * logf(xc) - (1.0f - y) * (1.0f - x1) * logf(x1);
        bool p  = x > 0.5f;
        bool yt = y > 0.5f;
        tp += ( p &&  yt) ? 1.0f : 0.0f;
        fp += ( p && !yt) ? 1.0f : 0.0f;
        fn += (!p &&  yt) ? 1.0f : 0.0f;
    }

    // ---- channels 1-2 (size), 3-4 (offset): y * |o - l| ----
    float szl = 0.f, ofl = 0.f;
    #pragma unroll
    for (int c = 1; c <= 4; ++c) {
        v4f oc = __builtin_nontemporal_load(&o4[base4 + (size_t)c * PLANE4]);
        v4f lc = __builtin_nontemporal_load(&l4[base4 + (size_t)c * PLANE4]);
        float acc = 0.f;
        #pragma unroll
        for (int j = 0; j < 4; ++j) acc += y4[j] * fabsf(oc[j] - lc[j]);
        if (c <= 2) szl += acc; else ofl += acc;
    }

    // ---- channels 5-24: fused argmax(labels) + gather(outputs) ----
    v4f best = __builtin_nontemporal_load(&l4[base4 + (size_t)5 * PLANE4]);
    v4f sel  = __builtin_nontemporal_load(&o4[base4 + (size_t)5 * PLANE4]);
    for (int c = 6; c < NC; ++c) {
        v4f lc = __builtin_nontemporal_load(&l4[base4 + (size_t)c * PLANE4]);
        v4f oc = __builtin_nontemporal_load(&o4[base4 + (size_t)c * PLANE4]);
        #pragma unroll
        for (int j = 0; j < 4; ++j) {
            if (lc[j] > best[j]) { best[j] = lc[j]; sel[j] = oc[j]; }  // strict '>' == first-max
        }
    }
    float cls = -(y4[0] * sel[0] + y4[1] * sel[1] + y4[2] * sel[2] + y4[3] * sel[3]);

    // ---- wave reduction via WMMA (EXEC all-ones here: uniform control flow) ----
    float q[7] = { obj, szl, ofl, cls, tp, fp, fn };
    float r[7];
    #pragma unroll
    for (int i = 0; i < 7; ++i) r[i] = wave_sum32(q[i]);

    __shared__ float red[8][8];
    const int wave = threadIdx.x >> 5;
    const int lane = threadIdx.x & 31;
    if (lane == 0) {
        #pragma unroll
        for (int i = 0; i < 7; ++i) red[wave][i] = r[i];
    }
    __syncthreads();

    if (threadIdx.x == 0) {
        float t[7] = {0.f, 0.f, 0.f, 0.f, 0.f, 0.f, 0.f};
        for (int w = 0; w < 8; ++w)
            #pragma unroll
            for (int i = 0; i < 7; ++i) t[i] += red[w][i];
        const int bid = blockIdx.y * 64 + blockIdx.x;     // 0..1023
        #pragma unroll
        for (int i = 0; i < 7; ++i) ws[(size_t)bid * 8 + i] = t[i];
    }
}

__global__ __launch_bounds__(256)
void yolo_final(const float* __restrict__ ws, float* __restrict__ out) {
    __shared__ float sA[256][4];
    __shared__ float sF[16];
    const int t = threadIdx.x;

    // deterministic sum of 1024 block records for the 4 loss partials
    float p[4] = {0.f, 0.f, 0.f, 0.f};
    for (int k = 0; k < 4; ++k) {
        const float* r = ws + (size_t)(t + 256 * k) * 8;
        #pragma unroll
        for (int i = 0; i < 4; ++i) p[i] += r[i];
    }
    #pragma unroll
    for (int i = 0; i < 4; ++i) sA[t][i] = p[i];
    __syncthreads();
    for (int s = 128; s > 0; s >>= 1) {
        if (t < s) {
            #pragma unroll
            for (int i = 0; i < 4; ++i) sA[t][i] += sA[t + s][i];
        }
        __syncthreads();
    }

    // per-batch F1 (lanes 0-15, fixed-order sums over the batch's 64 blocks)
    if (t < 16) {
        float tp = 0.f, fp = 0.f, fn = 0.f;
        for (int k = 0; k < 64; ++k) {
            const float* r = ws + (size_t)(t * 64 + k) * 8;
            tp += r[4]; fp += r[5]; fn += r[6];
        }
        float denom = 2.0f * tp + fp + fn;
        sF[t] = (denom > 0.0f) ? (2.0f * tp / fmaxf(denom, 1.0f)) : 0.0f;
    }
    __syncthreads();

    if (t == 0) {
        float obj = sA[0][0];
        float szl = 0.1f * sA[0][1];   // SIZE_SCALE
        float ofl = 0.1f * sA[0][2];   // OFFSET_SCALE
        float cll = 1.0f * sA[0][3];   // CLASS_SCALE
        float f1 = 0.f;
        for (int k = 0; k < 16; ++k) f1 += sF[k];
        f1 *= (1.0f / 16.0f);
        out[0] = obj + szl + ofl + cll;
        out[1] = f1;
        out[2] = obj;
        out[3] = szl;
        out[4] = ofl;
        out[5] = cll;
    }
}

extern "C" void kernel_launch(void* const* d_in, const int* in_sizes, int n_in,
                              void* d_out, int out_size, void* d_ws, size_t ws_size,
                              hipStream_t stream) {
    const float* outputs = (const float*)d_in[0];
    const float* labels  = (const float*)d_in[1];
    float* wsf = (float*)d_ws;   // uses 1024*8*4 = 32 KB
    float* out = (float*)d_out;  // 6 fp32 scalars

    dim3 grid(64, NB);           // 64 blocks per batch * 16 batches = 1024 blocks
    yolo_main<<<grid, 256, 0, stream>>>(outputs, labels, wsf);
    yolo_final<<<1, 256, 0, stream>>>(wsf, out);
}